// GraphAttentionLayer_32298154066154
// MI455X (gfx1250) — compile-verified
//
#include <hip/hip_runtime.h>
#include <math.h>

typedef __attribute__((ext_vector_type(2))) float v2f;
typedef __attribute__((ext_vector_type(8))) float v8f;
typedef __attribute__((ext_vector_type(4))) unsigned int v4u;
typedef __attribute__((ext_vector_type(8))) int v8i;
typedef __attribute__((ext_vector_type(4))) int v4i;

#define B_      8
#define N_      2048
#define INC     256
#define HEADS   4
#define OUTC    128
#define NE      65536
#define BN_EPS  1e-5f
#define NEG_SLOPE 0.2f
#define NEG_INF (-__builtin_inff())

// ---------------------------------------------------------------------------
// TDM helper: async-DMA a 16-row x 256-float tile (row pitch = stride elems)
// from global memory into LDS, padding +4 DWORDs every 256 DWORDs so the LDS
// image has a 260-float row stride (bank-conflict-free column reads).
// D# packing per CDNA5 ISA §8.3 (group 0) / §8.4 (group 1); groups 2/3 unused
// (2D tile). Issued by one wave; tracked on TENSORcnt.
// This toolchain exposes the 6-arg builtin (extra int32x8 group, zeroed).
// ---------------------------------------------------------------------------
__device__ __forceinline__ void tdm_load_tile16x256(const float* gsrc,
                                                    unsigned int lds_byte_addr,
                                                    unsigned int row_stride_elems) {
    unsigned long long ga = (unsigned long long)(size_t)gsrc;
    v4u g0;
    g0[0] = 1u;                                              // count=1 (valid user D#)
    g0[1] = lds_byte_addr;                                   // lds_addr (bytes)
    g0[2] = (unsigned int)ga;                                // global_addr[31:0]
    g0[3] = (unsigned int)((ga >> 32) & 0x01FFFFFFu)         // global_addr[56:32]
          | (2u << 30);                                      // type = 2 ("image")
    v8i g1;
    g1[0] = (int)((2u << 16)                                 // data_size = 4 bytes
                | (1u << 20)                                 // pad_enable
                | (7u << 22)                                 // pad_interval: 256 DWORDs
                | (3u << 25));                               // pad_amount:   4 DWORDs
    g1[1] = (int)((row_stride_elems & 0xFFFFu) << 16);       // tensor_dim0[15:0]
    g1[2] = (int)(((row_stride_elems >> 16) & 0xFFFFu)       // tensor_dim0[31:16]
                | (16u << 16));                              // tensor_dim1[15:0] = 16 rows
    g1[3] = (int)(256u << 16);                               // tensor_dim1 hi=0 | tile_dim0=256
    g1[4] = (int)16u;                                        // tile_dim1=16 | tile_dim2=0
    g1[5] = (int)row_stride_elems;                           // tensor_dim0_stride[31:0]
    g1[6] = 0;                                               // stride0 hi | stride1 lo
    g1[7] = 0;                                               // stride1 hi
    v4i z4 = {};
    v8i z8 = {};
    __builtin_amdgcn_tensor_load_to_lds(g0, g1, z4, z4, z8, 0);
}

// ---------------------------------------------------------------------------
// K0: fold head-mean and attention vector into the weights.
// ---------------------------------------------------------------------------
__global__ void prep_weights(const float* __restrict__ W, const float* __restrict__ a,
                             float* __restrict__ Wbar, float* __restrict__ wsd) {
    int k = blockIdx.x;      // input channel 0..255
    int c = threadIdx.x;     // 0..127
    __shared__ float red[128];
    const float* Wrow = W + (size_t)k * (HEADS * OUTC);
    float wb = 0.f;
    float ps[HEADS], pd[HEADS];
#pragma unroll
    for (int h = 0; h < HEADS; ++h) {
        float w = Wrow[h * OUTC + c];
        wb += w;
        ps[h] = w * a[c];
        pd[h] = w * a[OUTC + c];
    }
    Wbar[(size_t)k * OUTC + c] = wb * (1.0f / HEADS);
    for (int h = 0; h < HEADS; ++h) {
        red[c] = ps[h]; __syncthreads();
        for (int s = 64; s > 0; s >>= 1) { if (c < s) red[c] += red[c + s]; __syncthreads(); }
        if (c == 0) wsd[k * 8 + h] = red[0];
        __syncthreads();
        red[c] = pd[h]; __syncthreads();
        for (int s = 64; s > 0; s >>= 1) { if (c < s) red[c] += red[c + s]; __syncthreads(); }
        if (c == 0) wsd[k * 8 + 4 + h] = red[0];
        __syncthreads();
    }
}

// ---------------------------------------------------------------------------
// K1: hbar = X @ Wbar   (M=16384, K=256, N=128)  fp32 WMMA 16x16x4.
// A tile (16x256, contiguous rows) staged by the Tensor Data Mover.
// ---------------------------------------------------------------------------
__global__ void __launch_bounds__(256)
gemm_hbar(const float* __restrict__ X, const float* __restrict__ Wbar,
          float* __restrict__ hbar) {
    __shared__ float Alds[16][260];           // 260 = TDM-padded row pitch
    int row0 = blockIdx.x * 16;
    int tid  = threadIdx.x;
    if (tid < 32) {                           // wave 0 programs the TDM
        tdm_load_tile16x256(X + (size_t)row0 * INC,
                            (unsigned int)(size_t)(void*)&Alds[0][0], INC);
        __builtin_amdgcn_s_wait_tensorcnt(0);
    }
    __syncthreads();
    int wave  = tid >> 5, lane = tid & 31;
    int col0  = wave * 16;
    int ln15  = lane & 15;                    // A row (m) == B col (n) for this lane
    int khalf = (lane >> 4) * 2;              // lane-half K offset
    v8f acc = {};
#pragma unroll 4
    for (int k0 = 0; k0 < INC; k0 += 4) {
        v2f av, bv;
        av.x = Alds[ln15][k0 + khalf];
        av.y = Alds[ln15][k0 + khalf + 1];
        const float* bp = Wbar + (size_t)(k0 + khalf) * OUTC + col0 + ln15;
        bv.x = bp[0];
        bv.y = bp[OUTC];
        acc = __builtin_amdgcn_wmma_f32_16x16x4_f32(false, av, false, bv,
                                                    (short)0, acc, false, false);
    }
    float* out = hbar + (size_t)row0 * OUTC + col0 + ln15;
    int mbase = (lane >> 4) * 8;
#pragma unroll
    for (int i = 0; i < 8; ++i)
        out[(size_t)(mbase + i) * OUTC] = acc[i];
}

// ---------------------------------------------------------------------------
// K2: per-node attention scores s[node*8 + j] = X[node,:] . wsd[:,j]
// ---------------------------------------------------------------------------
__global__ void node_scores(const float* __restrict__ X, const float* __restrict__ wsd,
                            float* __restrict__ s) {
    int idx = blockIdx.x * blockDim.x + threadIdx.x;
    if (idx >= B_ * N_ * 8) return;
    int node = idx >> 3, j = idx & 7;
    const float* xr = X + (size_t)node * INC;
    float acc = 0.f;
    for (int k = 0; k < INC; ++k) acc += xr[k] * wsd[k * 8 + j];
    s[idx] = acc;
}

// ---------------------------------------------------------------------------
// K3: e_avg[b,e] = mean_h leaky_relu(s_src[b,src[e],h] + s_dst[b,dst[e],h])
// ---------------------------------------------------------------------------
__global__ void edge_scores(const long long* __restrict__ ei, const float* __restrict__ s,
                            float* __restrict__ eavg) {
    int idx = blockIdx.x * blockDim.x + threadIdx.x;
    if (idx >= B_ * NE) return;
    int b = idx / NE, e = idx % NE;
    int srcn = (int)ei[e];
    int dstn = (int)ei[NE + e];
    const float* ss = s + ((size_t)b * N_ + srcn) * 8;
    const float* sd = s + ((size_t)b * N_ + dstn) * 8 + 4;
    float acc = 0.f;
#pragma unroll
    for (int h = 0; h < 4; ++h) {
        float v = ss[h] + sd[h];
        acc += (v > 0.f) ? v : NEG_SLOPE * v;
    }
    eavg[idx] = acc * 0.25f;
}

// K4: attn[:] = -inf
__global__ void fill_neginf(float* __restrict__ attn, size_t n) {
    size_t i = (size_t)blockIdx.x * blockDim.x + threadIdx.x;
    size_t stride = (size_t)gridDim.x * blockDim.x;
    for (; i < n; i += stride) attn[i] = NEG_INF;
}

// K5: attn[b, dst[e], src[e]] = e_avg[b,e]   (overwrite semantics on dupes)
__global__ void scatter_edges(const long long* __restrict__ ei, const float* __restrict__ eavg,
                              float* __restrict__ attn) {
    int idx = blockIdx.x * blockDim.x + threadIdx.x;
    if (idx >= B_ * NE) return;
    int b = idx / NE, e = idx % NE;
    int srcn = (int)ei[e];
    int dstn = (int)ei[NE + e];
    attn[((size_t)b * N_ + dstn) * N_ + srcn] = eavg[idx];
}

// K6: row softmax over src axis; all -inf row -> zeros (nan_to_num)
__global__ void __launch_bounds__(256)
row_softmax(float* __restrict__ attn) {
    __shared__ float red[256];
    float* r = attn + (size_t)blockIdx.x * N_;
    int t = threadIdx.x;
    float mx = NEG_INF;
    for (int i = t; i < N_; i += 256) mx = fmaxf(mx, r[i]);
    red[t] = mx; __syncthreads();
    for (int s = 128; s > 0; s >>= 1) { if (t < s) red[t] = fmaxf(red[t], red[t + s]); __syncthreads(); }
    mx = red[0]; __syncthreads();
    if (mx == NEG_INF) {                       // row with no incoming edges
        for (int i = t; i < N_; i += 256) r[i] = 0.f;
        return;
    }
    float sum = 0.f;
    for (int i = t; i < N_; i += 256) { float v = expf(r[i] - mx); r[i] = v; sum += v; }
    red[t] = sum; __syncthreads();
    for (int s = 128; s > 0; s >>= 1) { if (t < s) red[t] += red[t + s]; __syncthreads(); }
    float inv = 1.f / red[0];
    for (int i = t; i < N_; i += 256) r[i] *= inv;
}

// ---------------------------------------------------------------------------
// K7: hout[b] = attn[b] @ hbar[b]   (M=2048, K=2048, N=128)  fp32 WMMA.
// A staged in 16x256 chunks by the TDM (strided rows, pitch N_); while wave 0
// programs the DMA, the other waves prefetch the next chunk toward L2.
// ---------------------------------------------------------------------------
__global__ void __launch_bounds__(256)
gemm_attn(const float* __restrict__ attn, const float* __restrict__ hbar,
          float* __restrict__ hout) {
    __shared__ float Alds[16][260];
    int b    = blockIdx.y;
    int row0 = blockIdx.x * 16;
    const float* A  = attn + (size_t)b * N_ * N_ + (size_t)row0 * N_;
    const float* Bm = hbar + (size_t)b * N_ * OUTC;
    int tid = threadIdx.x, wave = tid >> 5, lane = tid & 31;
    int col0  = wave * 16;
    int ln15  = lane & 15;
    int khalf = (lane >> 4) * 2;
    unsigned int lds_base = (unsigned int)(size_t)(void*)&Alds[0][0];
    v8f acc = {};
    for (int kb = 0; kb < N_; kb += 256) {
        __syncthreads();
        if (tid < 32) {                        // wave 0: DMA this A chunk into LDS
            tdm_load_tile16x256(A + kb, lds_base, N_);
            __builtin_amdgcn_s_wait_tensorcnt(0);
        } else if (kb + 256 < N_) {            // other waves: warm L2 for next chunk
            int i = tid - 32;
            if (i < 16)
                __builtin_prefetch(A + (size_t)i * N_ + kb + 256, 0, 1);
        }
        __syncthreads();
#pragma unroll 4
        for (int kk = 0; kk < 256; kk += 4) {
            v2f av, bv;
            av.x = Alds[ln15][kk + khalf];
            av.y = Alds[ln15][kk + khalf + 1];
            const float* bp = Bm + (size_t)(kb + kk + khalf) * OUTC + col0 + ln15;
            bv.x = bp[0];
            bv.y = bp[OUTC];
            acc = __builtin_amdgcn_wmma_f32_16x16x4_f32(false, av, false, bv,
                                                        (short)0, acc, false, false);
        }
    }
    float* out = hout + ((size_t)b * N_ + row0) * OUTC + col0 + ln15;
    int mbase = (lane >> 4) * 8;
#pragma unroll
    for (int i = 0; i < 8; ++i)
        out[(size_t)(mbase + i) * OUTC] = acc[i];
}

// K8: per-channel sum / sumsq (stat[0..127]=sum, stat[128..255]=sumsq)
__global__ void __launch_bounds__(256)
bn_stats(const float* __restrict__ hout, float* __restrict__ stat) {
    __shared__ float lsum[128], lsq[128];
    int t = threadIdx.x;
    if (t < 128) { lsum[t] = 0.f; lsq[t] = 0.f; }
    __syncthreads();
    int row0 = blockIdx.x * 64;
    int c = t & 127, r0 = t >> 7;
    float s0 = 0.f, s1 = 0.f;
    for (int r = r0; r < 64; r += 2) {
        float v = hout[(size_t)(row0 + r) * OUTC + c];
        s0 += v; s1 += v * v;
    }
    atomicAdd(&lsum[c], s0); atomicAdd(&lsq[c], s1);
    __syncthreads();
    if (t < 128) { atomicAdd(&stat[t], lsum[t]); atomicAdd(&stat[128 + t], lsq[t]); }
}

// K9: BatchNorm (batch stats, biased var) + ELU
__global__ void bn_elu(const float* __restrict__ hout, const float* __restrict__ stat,
                       const float* __restrict__ gamma, const float* __restrict__ beta,
                       float* __restrict__ out) {
    size_t i = (size_t)blockIdx.x * blockDim.x + threadIdx.x;
    if (i >= (size_t)B_ * N_ * OUTC) return;
    int c = (int)(i & (OUTC - 1));
    const float inv_n = 1.0f / (float)(B_ * N_);
    float mean = stat[c] * inv_n;
    float var  = stat[128 + c] * inv_n - mean * mean;
    float x = (hout[i] - mean) * rsqrtf(var + BN_EPS) * gamma[c] + beta[c];
    out[i] = x > 0.f ? x : (expf(x) - 1.f);
}

// ---------------------------------------------------------------------------
extern "C" void kernel_launch(void* const* d_in, const int* in_sizes, int n_in,
                              void* d_out, int out_size, void* d_ws, size_t ws_size,
                              hipStream_t stream) {
    const float*     x     = (const float*)d_in[0];
    const long long* ei    = (const long long*)d_in[1];   // int64 [2,E]
    const float*     W     = (const float*)d_in[2];
    const float*     a     = (const float*)d_in[3];
    const float*     gamma = (const float*)d_in[4];
    const float*     beta  = (const float*)d_in[5];
    float*           out   = (float*)d_out;

    float* ws = (float*)d_ws;
    size_t o = 0;
    float* Wbar = ws + o; o += (size_t)INC * OUTC;        // 32768
    float* wsd  = ws + o; o += (size_t)INC * 8;           // 2048
    float* hbar = ws + o; o += (size_t)B_ * N_ * OUTC;    // 2.10M
    float* s    = ws + o; o += (size_t)B_ * N_ * 8;       // 131072
    float* eavg = ws + o; o += (size_t)B_ * NE;           // 524288
    float* hout = ws + o; o += (size_t)B_ * N_ * OUTC;    // 2.10M
    float* stat = ws + o; o += 256;
    float* attn = ws + o; o += (size_t)B_ * N_ * N_;      // 33.55M (~134 MB, L2-resident)

    prep_weights <<<INC, 128, 0, stream>>>(W, a, Wbar, wsd);
    gemm_hbar    <<<(B_ * N_) / 16, 256, 0, stream>>>(x, Wbar, hbar);
    node_scores  <<<(B_ * N_ * 8) / 256, 256, 0, stream>>>(x, wsd, s);
    edge_scores  <<<(B_ * NE) / 256, 256, 0, stream>>>(ei, s, eavg);
    fill_neginf  <<<2048, 256, 0, stream>>>(attn, (size_t)B_ * N_ * N_);
    scatter_edges<<<(B_ * NE) / 256, 256, 0, stream>>>(ei, eavg, attn);
    row_softmax  <<<B_ * N_, 256, 0, stream>>>(attn);
    (void)hipMemsetAsync(stat, 0, 256 * sizeof(float), stream);
    gemm_attn    <<<dim3(N_ / 16, B_), 256, 0, stream>>>(attn, hbar, hout);
    bn_stats     <<<(B_ * N_) / 64, 256, 0, stream>>>(hout, stat);
    bn_elu       <<<(B_ * N_ * OUTC) / 256, 256, 0, stream>>>(hout, stat, gamma, beta, out);
}